// NutritionLoss_55482387530374
// MI455X (gfx1250) — compile-verified
//
#include <hip/hip_runtime.h>

typedef float v2f __attribute__((ext_vector_type(2)));
typedef float v8f __attribute__((ext_vector_type(8)));

#define FDIM   210    // 7*3*10 elements per batch row
#define KDIM   224    // 223 bins padded to multiple of 4 (WMMA K)
#define KPAD   225    // LDS row stride (odd vs 64 banks -> conflict-free A reads)
#define NB     16     // batches per workgroup
#define NCOPY  4      // f-split sub-histogram copies per stream
#define DRAW   1115   // 223*5 raw table dwords

__global__ __launch_bounds__(256) void nutrition_stage1(
    const float* __restrict__ pred_ids,
    const float* __restrict__ pred_amounts,
    const float* __restrict__ amounts,
    const float* __restrict__ data,
    const int*   __restrict__ ids,
    float* __restrict__ partial,
    int B)
{
  // [stream][copy][row][KPAD] : stream 0 = gold, 1 = pred
  __shared__ float hist[2 * NCOPY * NB * KPAD];
  __shared__ float rawD[1120];            // raw 223x5 table (async-loaded)
  __shared__ float red[8 * 8 * 32];       // per-wave partial C tiles

  const int t    = threadIdx.x;
  const int wave = t >> 5;
  const int lane = t & 31;
  const int b0   = blockIdx.x * NB;

  // ---- Kick off async DMA of the nutrient table into LDS (ASYNCcnt) ----
  {
    const unsigned long long gbase = (unsigned long long)data;
    for (int i = t; i < DRAW; i += 256) {
      unsigned ldst = (unsigned)(uintptr_t)(&rawD[i]);  // low 32 bits = LDS addr
      unsigned goff = (unsigned)i * 4u;
      asm volatile("global_load_async_to_lds_b32 %0, %1, %2"
                   :: "v"(ldst), "v"(goff), "s"(gbase) : "memory");
    }
  }

  // zero histograms while the DMA flies
  for (int i = t; i < 2 * NCOPY * NB * KPAD; i += 256) hist[i] = 0.f;

  asm volatile("s_wait_asynccnt 0x0" ::: "memory");
  __syncthreads();

  // ---- Phase 1: deterministic histograms, 8 waves x 16 rows, f split 4x ----
  // wave w: stream = w>>2, copy = w&3; lanes 0-15 own one LDS row each.
  if (lane < 16) {
    const int s    = wave >> 2;
    const int c    = wave & 3;
    const int row  = lane;
    const int b    = b0 + row;
    if (b < B) {
      const int f0 = (c * FDIM) / NCOPY;
      const int f1 = ((c + 1) * FDIM) / NCOPY;
      const long base = (long)b * FDIM;
      float* h = &hist[((s * NCOPY + c) * NB + row) * KPAD];
      if (s == 0) {                                  // gold: ids / amounts
        for (int f = f0; f < f1; ++f) {
          int k = ids[base + f];                     // 0..222
          h[k] += amounts[base + f];
        }
      } else {                                       // pred: round + bound
        for (int f = f0; f < f1; ++f) {
          float r = rintf(pred_ids[base + f]);       // round half-to-even
          int k = (r <= 222.f) ? (int)r : 0;         // id>222 -> ~0 -> row 0
          h[k] += pred_amounts[base + f];
        }
      }
    }
  }
  __syncthreads();

  // ---- merge the 4 f-copies into copy 0 of each stream (fixed order) ----
  for (int i = t; i < 2 * NB * KPAD; i += 256) {
    int s = i / (NB * KPAD), r = i % (NB * KPAD);
    float* dst = &hist[s * NCOPY * NB * KPAD + r];
    dst[0] += dst[1 * NB * KPAD] + dst[2 * NB * KPAD] + dst[3 * NB * KPAD];
  }
  __syncthreads();

  // ---- Phase 2: WMMA GEMM  H[16x224] * data_pad[224x16], K split 4 ways per stream ----
  // waves 0-3: gold K-chunks of 56; waves 4-7: pred.
  {
    const float* h = &hist[(wave < 4 ? 0 : NCOPY * NB * KPAD)];
    const int kbeg = (wave & 3) * 56;
    const int m    = lane & 15;
    const int koff = (lane >> 4) << 1;   // A/B frag: lanes>=16 carry K+2, K+3
    const int n    = lane & 15;

    // preload this wave's 14 B-fragments from the raw table (masked pad)
    float bx[14], by[14];
#pragma unroll
    for (int j = 0; j < 14; ++j) {
      int kx = kbeg + 4 * j + koff;
      bx[j] = (n < 5 && kx     < 223) ? rawD[kx * 5 + n]       : 0.f;
      by[j] = (n < 5 && kx + 1 < 223) ? rawD[(kx + 1) * 5 + n] : 0.f;
    }

    v8f cacc = {};
#pragma unroll
    for (int j = 0; j < 14; ++j) {
      int k0 = kbeg + 4 * j;
      v2f a, bm;
      a.x  = h[m * KPAD + k0 + koff];
      a.y  = h[m * KPAD + k0 + koff + 1];
      bm.x = bx[j];
      bm.y = by[j];
      cacc = __builtin_amdgcn_wmma_f32_16x16x4_f32(false, a, false, bm,
                                                   (short)0, cacc, false, false);
    }
#pragma unroll
    for (int r = 0; r < 8; ++r) red[wave * 256 + r * 32 + lane] = cacc[r];
  }
  __syncthreads();

  // ---- Phase 3: combine K-split tiles, masked |pred-gold|, wave reduce ----
  if (wave == 0) {
    float acc = 0.f;
#pragma unroll
    for (int r = 0; r < 8; ++r) {        // C/D layout: (M=r | r+8, N=lane&15)
      float g = 0.f, p = 0.f;
#pragma unroll
      for (int w = 0; w < 4; ++w) {
        g += red[w * 256 + r * 32 + lane];
        p += red[(w + 4) * 256 + r * 32 + lane];
      }
      if ((lane & 15) < 5) acc += fabsf(p - g);     // only 5 real nutrients
    }
#pragma unroll
    for (int off = 16; off > 0; off >>= 1) acc += __shfl_xor(acc, off, 32);
    if (lane == 0) partial[blockIdx.x] = acc;
  }
}

__global__ __launch_bounds__(256) void nutrition_stage2(
    const float* __restrict__ partial, float* __restrict__ out,
    int nblk, float scale)
{
  __shared__ float s[8];
  const int t = threadIdx.x;
  float acc = 0.f;
  for (int i = t; i < nblk; i += 256) acc += partial[i];
  for (int off = 16; off > 0; off >>= 1) acc += __shfl_xor(acc, off, 32);
  if ((t & 31) == 0) s[t >> 5] = acc;
  __syncthreads();
  if (t == 0) {
    float tot = 0.f;
    for (int w = 0; w < 8; ++w) tot += s[w];
    out[0] = tot * scale;                // scale = 1 / (100 * 7 * B)
  }
}

extern "C" void kernel_launch(void* const* d_in, const int* in_sizes, int n_in,
                              void* d_out, int out_size, void* d_ws, size_t ws_size,
                              hipStream_t stream) {
  const float* pred_ids     = (const float*)d_in[0];
  const float* pred_amounts = (const float*)d_in[1];
  const float* amounts      = (const float*)d_in[2];
  const float* data         = (const float*)d_in[3];
  const int*   ids          = (const int*)d_in[4];

  const int BF   = in_sizes[0];          // B * 210
  const int B    = BF / FDIM;            // 2048
  const int nblk = (B + NB - 1) / NB;    // 128 workgroups of 16 batches

  float* partial = (float*)d_ws;         // nblk floats of scratch

  nutrition_stage1<<<nblk, 256, 0, stream>>>(pred_ids, pred_amounts, amounts,
                                             data, ids, partial, B);
  const float scale = 1.0f / (700.0f * (float)B);
  nutrition_stage2<<<1, 256, 0, stream>>>(partial, (float*)d_out, nblk, scale);
}